// DFTD3_78022375899383
// MI455X (gfx1250) — compile-verified
//
#include <hip/hip_runtime.h>
#include <math.h>

// ---- D3 constants ----
#define D3_K1   16.0f
#define D3_K2   (4.0f/3.0f)
#define D3_K3   4.0f
#define D3_A1   0.4f
#define D3_A2   5.0f
#define D3_S6   1.0f
#define D3_S8   0.78f
#define CN_CUT  25.0f
#define DISP_CUT 50.0f
#define NREF    5
#define MAXZ_PAD 128   // LDS staging bound (actual MAX_Z = 95)

typedef float v2f __attribute__((ext_vector_type(2)));
typedef float v8f __attribute__((ext_vector_type(8)));

// Fast hardware reciprocal / sqrt (v_rcp_f32 / v_sqrt_f32, ~1 ulp)
__device__ __forceinline__ float frcp(float x)  { return __builtin_amdgcn_rcpf(x); }
__device__ __forceinline__ float fsqrt(float x) { return __builtin_amdgcn_sqrtf(x); }

// --------------------------------------------------------------------------
// Wave32 sum via V_WMMA_F32_16X16X4_F32:
//  A (16x4 f32, 2 VGPRs): VGPR0 lane L -> (M=L%16, K = L<16 ? 0 : 2), VGPR1 = 0
//  B (4x16) = all ones, C = 0  =>  D[m,n] = p_m + p_{m+16}
//  Per-lane sum of the 8 D VGPRs: lanes 0-15 hold sum(p0..7,p16..23),
//  lanes 16-31 hold sum(p8..15,p24..31); shfl_xor(16) completes the wave sum.
// --------------------------------------------------------------------------
__device__ __forceinline__ float wave_sum_wmma(float p) {
    v2f a; a.x = p;    a.y = 0.0f;
    v2f b; b.x = 1.0f; b.y = 1.0f;
    v8f c = {};
    v8f d = __builtin_amdgcn_wmma_f32_16x16x4_f32(
        /*neg_a=*/false, a, /*neg_b=*/false, b,
        /*c_mod=*/(short)0, c, /*reuse_a=*/false, /*reuse_b=*/false);
    float s = d[0] + d[1] + d[2] + d[3] + d[4] + d[5] + d[6] + d[7];
    s += __shfl_xor(s, 16, 32);
    return s;   // uniform across the wave
}

// CN-weighted Gaussian reference weights (matches _atom_weights incl. fallback)
__device__ __forceinline__ void atom_weights(float cn, const float* __restrict__ refs,
                                             float* __restrict__ w) {
    float gw[NREF];
    float norm = 0.0f;
    float best = -INFINITY;
    int   bidx = 0;
#pragma unroll
    for (int a = 0; a < NREF; ++a) {
        float r = refs[a];
        bool valid = (r >= 0.0f);
        float dc = cn - r;
        float g = valid ? __expf(-D3_K3 * dc * dc) : 0.0f;
        gw[a] = g;
        norm += g;
        if (valid && r > best) { best = r; bidx = a; }
    }
    if (norm > 1e-30f) {
        float inv = frcp(norm);
#pragma unroll
        for (int a = 0; a < NREF; ++a) w[a] = gw[a] * inv;
    } else {
#pragma unroll
        for (int a = 0; a < NREF; ++a) w[a] = (a == bidx) ? 1.0f : 0.0f;
    }
}

// --------------------------------------------------------------------------
__global__ void d3_zero_kernel(float* __restrict__ cn, float* __restrict__ out, int n) {
    int t = blockIdx.x * blockDim.x + threadIdx.x;
    if (t < n) cn[t] = 0.0f;
    if (t == 0) out[0] = 0.0f;
}

// Pass 1: coordination numbers (scatter-add with native fp32 atomics)
__global__ __launch_bounds__(256)
void d3_cn_kernel(const float* __restrict__ pos, const int* __restrict__ num,
                  const int* __restrict__ ei, const int* __restrict__ ej,
                  const float* __restrict__ rcov, float* __restrict__ cn,
                  int E, int maxz) {
    __shared__ float s_rcov[MAXZ_PAD];
    for (int t = threadIdx.x; t < maxz; t += blockDim.x) s_rcov[t] = rcov[t];
    __syncthreads();

    int stride = gridDim.x * blockDim.x;
    for (int e = blockIdx.x * blockDim.x + threadIdx.x; e < E; e += stride) {
        int i = ei[e], j = ej[e];
        float dx = pos[3 * j + 0] - pos[3 * i + 0];
        float dy = pos[3 * j + 1] - pos[3 * i + 1];
        float dz = pos[3 * j + 2] - pos[3 * i + 2];
        float d2 = dx * dx + dy * dy + dz * dz + 1e-20f;
        float d = fsqrt(d2);
        if (d < CN_CUT) {
            float rr = s_rcov[num[i]] + s_rcov[num[j]];
            float x = D3_K1 * (D3_K2 * rr * frcp(d) - 1.0f);
            float cp = frcp(1.0f + __expf(-x));   // sigmoid
            unsafeAtomicAdd(&cn[i], cp);          // global_atomic_add_f32 (no-return)
        }
    }
}

// Pass 2: C6 interpolation + BJ-damped energy, reduced to a scalar
__global__ __launch_bounds__(256)
void d3_energy_kernel(const float* __restrict__ pos, const int* __restrict__ num,
                      const int* __restrict__ ei, const int* __restrict__ ej,
                      const float* __restrict__ r4r2, const float* __restrict__ c6tab,
                      const float* __restrict__ cnref, const float* __restrict__ cn,
                      float* __restrict__ out, int E, int maxz) {
    __shared__ float s_cnref[MAXZ_PAD * NREF];
    __shared__ float s_r4r2[MAXZ_PAD];
    __shared__ float s_wsum[8];

    for (int t = threadIdx.x; t < maxz * NREF; t += blockDim.x) s_cnref[t] = cnref[t];
    for (int t = threadIdx.x; t < maxz; t += blockDim.x) s_r4r2[t] = r4r2[t];
    __syncthreads();

    float acc = 0.0f;
    int stride = gridDim.x * blockDim.x;
    for (int e = blockIdx.x * blockDim.x + threadIdx.x; e < E; e += stride) {
        int i = ei[e], j = ej[e];
        float dx = pos[3 * j + 0] - pos[3 * i + 0];
        float dy = pos[3 * j + 1] - pos[3 * i + 1];
        float dz = pos[3 * j + 2] - pos[3 * i + 2];
        float d2 = dx * dx + dy * dy + dz * dz + 1e-20f;
        float d = fsqrt(d2);
        if (d < DISP_CUT) {
            int zi = num[i], zj = num[j];
            float wi[NREF], wj[NREF];
            atom_weights(cn[i], &s_cnref[zi * NREF], wi);
            atom_weights(cn[j], &s_cnref[zj * NREF], wj);

            const float* C = &c6tab[((size_t)zi * maxz + zj) * (NREF * NREF)];
            float c6 = 0.0f;
#pragma unroll
            for (int a = 0; a < NREF; ++a) {
                float row = 0.0f;
#pragma unroll
                for (int b = 0; b < NREF; ++b)
                    row = fmaf(wj[b], C[a * NREF + b], row);
                c6 = fmaf(wi[a], row, c6);
            }
            float qq = 3.0f * s_r4r2[zi] * s_r4r2[zj];
            float c8 = c6 * qq;
            float f  = D3_A1 * fsqrt(qq) + D3_A2;
            float f2 = f * f;
            float f6 = f2 * f2 * f2;
            float f8 = f6 * f2;
            float d6 = d2 * d2 * d2;
            float d8 = d6 * d2;
            acc -= 0.5f * (D3_S6 * c6 * frcp(d6 + f6) + D3_S8 * c8 * frcp(d8 + f8));
        }
    }

    // intra-wave reduction on the WMMA pipe, then cross-wave via LDS
    float s = wave_sum_wmma(acc);
    int lane = threadIdx.x & 31;
    int wave = threadIdx.x >> 5;
    if (lane == 0) s_wsum[wave] = s;
    __syncthreads();
    if (threadIdx.x == 0) {
        float tot = 0.0f;
        int nw = blockDim.x >> 5;
        for (int w = 0; w < nw; ++w) tot += s_wsum[w];
        unsafeAtomicAdd(out, tot);
    }
}

// --------------------------------------------------------------------------
extern "C" void kernel_launch(void* const* d_in, const int* in_sizes, int n_in,
                              void* d_out, int out_size, void* d_ws, size_t ws_size,
                              hipStream_t stream) {
    const float* pos    = (const float*)d_in[0];
    const int*   num    = (const int*)  d_in[1];
    const int*   ei     = (const int*)  d_in[2];
    const int*   ej     = (const int*)  d_in[3];
    const float* rcov   = (const float*)d_in[4];
    const float* r4r2   = (const float*)d_in[5];
    const float* c6tab  = (const float*)d_in[6];
    const float* cnref  = (const float*)d_in[7];
    float*       out    = (float*)d_out;

    int N    = in_sizes[1];             // atoms
    int E    = in_sizes[2];             // directed edges
    int maxz = in_sizes[4];             // element table size (95)

    float* cn = (float*)d_ws;           // N floats of scratch

    d3_zero_kernel<<<(N + 255) / 256, 256, 0, stream>>>(cn, out, N);

    int blocks = (E + 255) / 256;
    if (blocks > 4096) blocks = 4096;
    d3_cn_kernel<<<blocks, 256, 0, stream>>>(pos, num, ei, ej, rcov, cn, E, maxz);
    d3_energy_kernel<<<blocks, 256, 0, stream>>>(pos, num, ei, ej, r4r2, c6tab,
                                                 cnref, cn, out, E, maxz);
}